// SubtitleRefinedL1BalanceCELoss_15204184228057
// MI455X (gfx1250) — compile-verified
//
#include <hip/hip_runtime.h>
#include <hip/hip_bf16.h>
#include <math.h>

// ---------------------------------------------------------------------------
// Problem geometry (fixed by the reference: N=16, H=640, W=640, fp32)
// ---------------------------------------------------------------------------
#define NN  16
#define HH  640
#define WW  640
#define HWE (HH * WW)      // 409600 (divisible by 4)
#define NPE (NN * HWE)     // 6553600 elements per array
#define NP4 (NPE / 4)      // 1638400 float4 groups
#define NCOL (NN * WW)     // 10240 columns total
#define NCG  (NCOL / 4)    // 2560 column groups

// Workspace layout in 32-bit words
#define OFF_ACC    0       // 8 f32 accumulators
#define OFF_SEL    8       // [0]=K u32 [1]=neg_count f32 [2]=Krem u32 [3]=prefix u32 [5]=S_gt f32
#define OFF_HIST   16      // 2048 u32 histogram bins
#define OFF_COUNTS 2064    // 10240 f32 per-column counts
#define OFF_CENTER 12304   // 10240 f32 per-column centers
#define OFF_KEEP   22544   // 16 f32 keep flags
// total = 22560 words = 90240 bytes of d_ws

typedef float v2f __attribute__((ext_vector_type(2)));
typedef float v4f __attribute__((ext_vector_type(4)));
typedef float v8f __attribute__((ext_vector_type(8)));

// ---------------------------------------------------------------------------
// Wave32 sum via V_WMMA_F32_16X16X4_F32 (XDL pipe, layout-agnostic):
// A = ones(16x4)  =>  D[m][n] = sum_k B[k][n]  (identical rows).
// Each lane puts its value in B slot0, 0 in slot1 => sum of all B slots =
// sum over lanes. D VGPR0 across 32 lanes covers exactly two full rows
// (documented C/D layout), so lane-sum(D[0]) = 2 * wave total.
// ---------------------------------------------------------------------------
__device__ __forceinline__ float wave_sum_f32(float x) {
    v2f a; a[0] = 1.0f; a[1] = 1.0f;
    v2f b; b[0] = x;    b[1] = 0.0f;
    v8f c = {0.f, 0.f, 0.f, 0.f, 0.f, 0.f, 0.f, 0.f};
    v8f d = __builtin_amdgcn_wmma_f32_16x16x4_f32(false, a, false, b,
                                                  (short)0, c, false, false);
    float s = d[0];
#pragma unroll
    for (int off = 16; off > 0; off >>= 1)
        s += __shfl_xor(s, off, 32);
    return 0.5f * s;
}

// Block (256 threads = 8 waves) reduction of 8 quantities.
// On exit threads 0..7 hold total of quantity threadIdx.x in q[0].
__device__ __forceinline__ void block_reduce8(float* q, float* lds) {
    const int lane = threadIdx.x & 31;
    const int wave = threadIdx.x >> 5;
    float s[8];
#pragma unroll
    for (int i = 0; i < 8; ++i) s[i] = wave_sum_f32(q[i]);
    if (lane == 0) {
#pragma unroll
        for (int i = 0; i < 8; ++i) lds[wave * 8 + i] = s[i];
    }
    __syncthreads();
    if (threadIdx.x < 8) {
        float t = 0.0f;
#pragma unroll
        for (int w = 0; w < 8; ++w) t += lds[w * 8 + (int)threadIdx.x];
        q[0] = t;
    }
    __syncthreads();
}

// Shared elementwise BCE loss (identical codegen across kernels matters for
// the radix-select bit comparisons).
__device__ __forceinline__ float bce_elem(float pb, float g) {
    float p = fminf(fmaxf(pb, 1e-7f), 1.0f - 1e-7f);
    return -(g * logf(p) + (1.0f - g) * log1pf(-p));
}

// loss*neg for one element (used identically by k_hist and k_sumtop).
__device__ __forceinline__ float neg_val_s(float tb, float m, float g,
                                           float pbv, float kp) {
    float b   = (tb > 0.5f) ? 1.0f : 0.0f;
    float cm  = (b * m * kp) * m;           // combined_mask
    float neg = (1.0f - g) * cm;
    return bce_elem(pbv, g) * neg;
}

// ---------------------------------------------------------------------------
// Kernels
// ---------------------------------------------------------------------------
__global__ __launch_bounds__(256) void k_zero(unsigned* wsu) {
    for (int i = threadIdx.x; i < OFF_COUNTS; i += 256) wsu[i] = 0u;
}

// Per-column text stats, 4 columns per thread with b128 row loads.
__global__ __launch_bounds__(256) void k_colstats(
        const float* __restrict__ ptb, const float* __restrict__ msk,
        float* __restrict__ counts, float* __restrict__ center) {
    int cg = blockIdx.x * 256 + threadIdx.x;
    if (cg >= NCG) return;
    int c0 = cg * 4;
    int n = c0 / WW, col = c0 - n * WW;     // 4-col group never crosses a sample
    const v4f* pt = (const v4f*)(ptb + (size_t)n * HWE + col);
    const v4f* mk = (const v4f*)(msk + (size_t)n * HWE + col);
    float cnt[4] = {0, 0, 0, 0};
    int ymin[4] = {HH, HH, HH, HH};
    int ymax[4] = {-1, -1, -1, -1};
    for (int y = 0; y < HH; ++y) {
        v4f tv = pt[(size_t)y * (WW / 4)];
        v4f mv = mk[(size_t)y * (WW / 4)];
#pragma unroll
        for (int j = 0; j < 4; ++j) {
            float b = (tv[j] > 0.5f) ? 1.0f : 0.0f;
            float t = b * mv[j];
            cnt[j] += t;
            if (t > 0.0f) {
                if (y < ymin[j]) ymin[j] = y;
                if (y > ymax[j]) ymax[j] = y;
            }
        }
    }
#pragma unroll
    for (int j = 0; j < 4; ++j) {
        counts[c0 + j] = cnt[j];
        center[c0 + j] = 0.5f * (float)(ymin[j] + ymax[j]);
    }
}

// Per-sample flatness decision (tilt CV + wobble std, Bessel-corrected).
__global__ __launch_bounds__(256) void k_keep(
        const float* __restrict__ counts, const float* __restrict__ center,
        float* __restrict__ keep) {
    __shared__ float lds[64];
    int n = blockIdx.x;
    float q[8] = {0, 0, 0, 0, 0, 0, 0, 0};
    for (int c = threadIdx.x; c < WW; c += 256) {
        float cnt = counts[n * WW + c];
        float act = (cnt > 0.0f) ? 1.0f : 0.0f;
        float ctr = center[n * WW + c] * act;
        q[0] += act;
        q[1] += cnt * act;
        q[2] += cnt * cnt * act;
        q[3] += ctr;
        q[4] += ctr * ctr * act;
    }
    block_reduce8(q, lds);
    if (threadIdx.x < 8) lds[threadIdx.x] = q[0];
    __syncthreads();
    if (threadIdx.x == 0) {
        float nact = lds[0], sc = lds[1], sc2 = lds[2], sct = lds[3], sct2 = lds[4];
        float kv = 0.0f;
        if (nact >= 2.0f) {
            float mean = sc / nact;
            float var  = fmaxf((sc2 - nact * mean * mean) / (nact - 1.0f), 0.0f);
            float cv   = sqrtf(var) / fmaxf(mean, 1e-6f);
            bool tilt  = (mean > 1e-6f) && (cv > 0.3f);
            float mc   = sct / nact;
            float varc = fmaxf((sct2 - nact * mc * mc) / (nact - 1.0f), 0.0f);
            bool wob   = sqrtf(varc) > 5.0f;
            kv = (!tilt && !wob) ? 1.0f : 0.0f;
        }
        keep[n] = kv;
    }
}

// Fused main pass: dice parts, L1 parts, pos/neg counts, pos BCE loss.
// b128 loads; re-read arrays use RT hints (stay in 192MB L2), single-use
// arrays (pred_thresh, thresh_map, thresh_mask) load non-temporally.
__global__ __launch_bounds__(256) void k_main(
        const float* __restrict__ pb,  const float* __restrict__ pt,
        const float* __restrict__ ptb, const float* __restrict__ gt,
        const float* __restrict__ msk, const float* __restrict__ tmap,
        const float* __restrict__ tmsk, const float* __restrict__ keep,
        float* __restrict__ acc) {
    __shared__ float lds[64];
    const v4f* pb4  = (const v4f*)pb;
    const v4f* pt4  = (const v4f*)pt;
    const v4f* ptb4 = (const v4f*)ptb;
    const v4f* gt4  = (const v4f*)gt;
    const v4f* msk4 = (const v4f*)msk;
    const v4f* tmp4 = (const v4f*)tmap;
    const v4f* tmk4 = (const v4f*)tmsk;
    float q[8] = {0, 0, 0, 0, 0, 0, 0, 0};
    int stride = gridDim.x * blockDim.x;
    for (int i4 = blockIdx.x * blockDim.x + threadIdx.x; i4 < NP4; i4 += stride) {
        int n = (i4 * 4) / HWE;             // all 4 elems in same sample
        float kp = keep[n];
        v4f tbv = ptb4[i4];
        v4f mv  = msk4[i4];
        v4f gv  = gt4[i4];
        v4f pbv = pb4[i4];
        v4f ptv = __builtin_nontemporal_load(&pt4[i4]);
        v4f tmv = __builtin_nontemporal_load(&tmp4[i4]);
        v4f tkv = __builtin_nontemporal_load(&tmk4[i4]);
#pragma unroll
        for (int j = 0; j < 4; ++j) {
            float tb = tbv[j], m = mv[j], g = gv[j];
            float b   = (tb > 0.5f) ? 1.0f : 0.0f;
            float sub = b * m * kp;
            float cm  = sub * m;
            float ctm = sub * tkv[j];
            q[0] += tb * g * cm;                    // dice inter
            q[1] += tb * cm;                        // dice p-sum
            q[2] += g * cm;                         // dice g-sum
            q[3] += fabsf(ptv[j] - tmv[j]) * ctm;   // l1 numerator
            q[4] += ctm;                            // l1 denominator
            float loss = bce_elem(pbv[j], g);
            float pos  = g * cm;
            q[5] += pos;                            // pos_count
            q[6] += (1.0f - g) * cm;                // neg_count (raw)
            q[7] += loss * pos;                     // pos_loss
        }
    }
    block_reduce8(q, lds);
    if (threadIdx.x < 8) atomicAdd(&acc[threadIdx.x], q[0]);
}

// K = ceil(min(neg_raw, 3*pos)); init selection state.
__global__ void k_prepK(const float* acc, unsigned* sel) {
    float pos = acc[5], negr = acc[6];
    float negc = fminf(negr, 3.0f * pos);
    float kc = ceilf(negc);
    unsigned K = (kc <= 0.0f) ? 0u :
                 (kc >= (float)NPE ? (unsigned)NPE : (unsigned)kc);
    sel[0] = K;
    ((float*)sel)[1] = negc;
    sel[2] = K;                 // Krem
    sel[3] = 0u;                // prefix
    sel[4] = 0u;
    ((float*)sel)[5] = 0.0f;    // S_gt
}

// Radix histogram pass (LDS-local bins, then merge). b128 loads; the four
// arrays touched here are the L2-resident working set (RT hints).
__global__ __launch_bounds__(256) void k_hist(
        const float* __restrict__ ptb, const float* __restrict__ msk,
        const float* __restrict__ gt,  const float* __restrict__ pb,
        const float* __restrict__ keep, const unsigned* __restrict__ sel,
        unsigned* __restrict__ hist, int shift, int binMask, int prefShift) {
    __shared__ unsigned lh[2048];
    if (sel[0] == 0u) return;
    unsigned prefix = sel[3];
    for (int i = threadIdx.x; i < 2048; i += 256) lh[i] = 0u;
    __syncthreads();
    const v4f* ptb4 = (const v4f*)ptb;
    const v4f* msk4 = (const v4f*)msk;
    const v4f* gt4  = (const v4f*)gt;
    const v4f* pb4  = (const v4f*)pb;
    int stride = gridDim.x * blockDim.x;
    for (int i4 = blockIdx.x * blockDim.x + threadIdx.x; i4 < NP4; i4 += stride) {
        float kp = keep[(i4 * 4) / HWE];
        v4f tbv = ptb4[i4];
        v4f mv  = msk4[i4];
        v4f gv  = gt4[i4];
        v4f pbv = pb4[i4];
#pragma unroll
        for (int j = 0; j < 4; ++j) {
            float v = neg_val_s(tbv[j], mv[j], gv[j], pbv[j], kp);
            unsigned u = __float_as_uint(v);
            bool ok = (prefShift >= 32) || ((u >> prefShift) == prefix);
            if (ok) atomicAdd(&lh[(u >> shift) & (unsigned)binMask], 1u);
        }
    }
    __syncthreads();
    for (int i = threadIdx.x; i < 2048; i += 256)
        if (lh[i]) atomicAdd(&hist[i], lh[i]);
}

// Single-thread scan: find the bin containing the Krem-th largest value,
// update prefix/Krem, clear histogram for the next pass.
__global__ void k_scan(unsigned* sel, unsigned* hist, int bitsThisPass, int nbins) {
    if (sel[0] != 0u) {
        unsigned Krem = sel[2], prefix = sel[3];
        unsigned above = 0u;
        int selb = 0;
        for (int b = nbins - 1; b >= 0; --b) {
            unsigned h = hist[b];
            if (above + h >= Krem) { selb = b; break; }
            above += h;
        }
        sel[2] = Krem - above;
        sel[3] = (prefix << bitsThisPass) | (unsigned)selb;
    }
    for (int b = 0; b < nbins; ++b) hist[b] = 0u;
}

// Sum of all values strictly above the exact bit threshold.
__global__ __launch_bounds__(256) void k_sumtop(
        const float* __restrict__ ptb, const float* __restrict__ msk,
        const float* __restrict__ gt,  const float* __restrict__ pb,
        const float* __restrict__ keep, const unsigned* __restrict__ sel,
        float* __restrict__ Sgt) {
    __shared__ float lds1[8];
    if (sel[0] == 0u) return;
    unsigned thr = sel[3];
    const v4f* ptb4 = (const v4f*)ptb;
    const v4f* msk4 = (const v4f*)msk;
    const v4f* gt4  = (const v4f*)gt;
    const v4f* pb4  = (const v4f*)pb;
    float part = 0.0f;
    int stride = gridDim.x * blockDim.x;
    for (int i4 = blockIdx.x * blockDim.x + threadIdx.x; i4 < NP4; i4 += stride) {
        float kp = keep[(i4 * 4) / HWE];
        v4f tbv = ptb4[i4];
        v4f mv  = msk4[i4];
        v4f gv  = gt4[i4];
        v4f pbv = pb4[i4];
#pragma unroll
        for (int j = 0; j < 4; ++j) {
            float v = neg_val_s(tbv[j], mv[j], gv[j], pbv[j], kp);
            if (__float_as_uint(v) > thr) part += v;
        }
    }
    float s = wave_sum_f32(part);
    if ((threadIdx.x & 31) == 0) lds1[threadIdx.x >> 5] = s;
    __syncthreads();
    if (threadIdx.x == 0) {
        float t = 0.0f;
#pragma unroll
        for (int w = 0; w < 8; ++w) t += lds1[w];
        atomicAdd(Sgt, t);
    }
}

__global__ void k_final(const float* acc, const unsigned* sel, float* out) {
    const float* selF = (const float*)sel;
    unsigned K = sel[0], Krem = sel[2];
    float thr = __uint_as_float(sel[3]);
    float neg_sum = (K == 0u) ? 0.0f : (selF[5] + (float)Krem * thr);
    float bce  = (acc[7] + neg_sum) / (acc[5] + selF[1] + 1e-6f);
    float l1   = acc[3] / fmaxf(acc[4], 1e-6f);
    float dice = 1.0f - 2.0f * acc[0] / (acc[1] + acc[2] + 1e-6f);
    out[0] = dice + 10.0f * l1 + 5.0f * bce;
}

// ---------------------------------------------------------------------------
// Launch
// ---------------------------------------------------------------------------
extern "C" void kernel_launch(void* const* d_in, const int* in_sizes, int n_in,
                              void* d_out, int out_size, void* d_ws, size_t ws_size,
                              hipStream_t stream) {
    const float* pb   = (const float*)d_in[0];  // pred_binary
    const float* pt   = (const float*)d_in[1];  // pred_thresh
    const float* ptb  = (const float*)d_in[2];  // pred_thresh_binary
    const float* gt   = (const float*)d_in[3];  // gt
    const float* msk  = (const float*)d_in[4];  // mask
    const float* tmap = (const float*)d_in[5];  // thresh_map
    const float* tmsk = (const float*)d_in[6];  // thresh_mask

    float*    wsf = (float*)d_ws;
    unsigned* wsu = (unsigned*)d_ws;
    float*    acc    = wsf + OFF_ACC;
    unsigned* sel    = wsu + OFF_SEL;
    unsigned* hist   = wsu + OFF_HIST;
    float*    counts = wsf + OFF_COUNTS;
    float*    center = wsf + OFF_CENTER;
    float*    keep   = wsf + OFF_KEEP;

    k_zero<<<1, 256, 0, stream>>>(wsu);
    k_colstats<<<(NCG + 255) / 256, 256, 0, stream>>>(ptb, msk, counts, center);
    k_keep<<<NN, 256, 0, stream>>>(counts, center, keep);
    k_main<<<2048, 256, 0, stream>>>(pb, pt, ptb, gt, msk, tmap, tmsk, keep, acc);
    k_prepK<<<1, 1, 0, stream>>>(acc, sel);
    // exact top-K radix select over bits [31:21], [20:10], [9:0]
    k_hist<<<512, 256, 0, stream>>>(ptb, msk, gt, pb, keep, sel, hist, 21, 2047, 32);
    k_scan<<<1, 1, 0, stream>>>(sel, hist, 11, 2048);
    k_hist<<<512, 256, 0, stream>>>(ptb, msk, gt, pb, keep, sel, hist, 10, 2047, 21);
    k_scan<<<1, 1, 0, stream>>>(sel, hist, 11, 2048);
    k_hist<<<512, 256, 0, stream>>>(ptb, msk, gt, pb, keep, sel, hist, 0, 1023, 10);
    k_scan<<<1, 1, 0, stream>>>(sel, hist, 10, 1024);
    k_sumtop<<<512, 256, 0, stream>>>(ptb, msk, gt, pb, keep, sel, (float*)(wsu + OFF_SEL + 5));
    k_final<<<1, 1, 0, stream>>>(acc, sel, (float*)d_out);
}